// FarthestSample_19645180412393
// MI455X (gfx1250) — compile-verified
//
#include <hip/hip_runtime.h>
#include <stdint.h>

#define NBATCH   32
#define NPTS     131072
#define NPOINTS  4096
#define WGPB     8                    // workgroups cooperating per batch
#define CHUNK    (NPTS / WGPB)        // 16384 points per WG
#define THREADS  256
#define TILE     512                  // points per staged tile
#define NTILES   (CHUNK / TILE)       // 32

struct Ctrl { int cnt; int pad0[15]; int gen; int pad1[15]; };  // 128B per batch

__device__ __forceinline__ uint32_t lds_byte_addr(const void* p) {
  // generic pointer to LDS: low 32 bits are the wave-relative LDS byte address
  return (uint32_t)(uintptr_t)p;
}

// Stage one 512-point tile (6144 B) into LDS with async global->LDS copies.
// Per wave: 768 B slice; each of the 3 b64 instructions is a fully coalesced
// 256 B burst (lane*8), offsets 0/256/512. INST_OFFSET is added to both the
// global and LDS addresses (ISA 8.2). Tracked by ASYNCcnt. Thread tid later
// consumes bytes [24*tid, 24*tid+24), which lie inside its own wave's slice,
// so a per-wave s_wait_asynccnt suffices (no workgroup barrier needed).
__device__ __forceinline__ void issue_tile_copy(const float* src, const float* ldsDst, int tid) {
  const uint32_t wv   = (uint32_t)(tid >> 5);
  const uint32_t lane = (uint32_t)(tid & 31);
  const uint32_t off  = wv * 768u + lane * 8u;
  uint64_t g = (uint64_t)(uintptr_t)src + off;
  uint32_t l = lds_byte_addr(ldsDst) + off;
  asm volatile(
      "global_load_async_to_lds_b64 %0, %1, off offset:0\n\t"
      "global_load_async_to_lds_b64 %0, %1, off offset:256\n\t"
      "global_load_async_to_lds_b64 %0, %1, off offset:512"
      :
      : "v"(l), "v"(g)
      : "memory");
}

__device__ __forceinline__ void wait_async_zero() {
  asm volatile("s_wait_asynccnt 0x0" ::: "memory");
}

__global__ void fps_init_kernel(Ctrl* ctrl, int* winner) {
  int t = threadIdx.x;
  if (t < NBATCH) {
    ctrl[t].cnt = 0;
    ctrl[t].gen = 0;
    winner[t] = 0;
  }
}

__global__ __launch_bounds__(THREADS, 1)
void fps_main_kernel(const float* __restrict__ inp, float* __restrict__ outPts,
                     int* __restrict__ outIdx, Ctrl* __restrict__ ctrl,
                     float* __restrict__ partV, int* __restrict__ partI,
                     int* __restrict__ winner) {
#pragma clang fp contract(off)
  __shared__ float dist[CHUNK];                           // 64 KB running min-sq-dist
  __shared__ __align__(16) float tileBuf[2][TILE * 3];    // 12 KB double-buffered points
  __shared__ float redV[THREADS / 32];
  __shared__ int   redI[THREADS / 32];
  __shared__ int   bcast;

  const int bx   = blockIdx.x;
  const int b    = bx >> 3;   // batch
  const int c    = bx & 7;    // chunk within batch
  const int tid  = threadIdx.x;
  const int lane = tid & 31;
  const int wv   = tid >> 5;

  const float* pts   = inp + (size_t)b * NPTS * 3;
  const float* mypts = pts + (size_t)c * CHUNK * 3;

  for (int i = tid; i < CHUNK; i += THREADS) dist[i] = 1e10f;

  if (c == 0 && tid == 0) {
    outIdx[(size_t)b * NPOINTS] = 0;                      // first selected index is 0
    float* op = outPts + (size_t)b * NPOINTS * 3;
    op[0] = pts[0]; op[1] = pts[1]; op[2] = pts[2];
  }
  __syncthreads();

  issue_tile_copy(mypts, tileBuf[0], tid);                // prologue: stage tile 0

  int w = 0;                                              // current winner (global idx in batch)
  for (int k = 1; k < NPOINTS; ++k) {
    const float qx = pts[3 * (size_t)w + 0];
    const float qy = pts[3 * (size_t)w + 1];
    const float qz = pts[3 * (size_t)w + 2];

    float bestV = -1.0f;
    int   bestI = 0;

    for (int t = 0; t < NTILES; ++t) {
      wait_async_zero();                                  // my wave's slice of tile t is in LDS
      if (t + 1 < NTILES)
        issue_tile_copy(mypts + (size_t)(t + 1) * TILE * 3, tileBuf[(t + 1) & 1], tid);

      const float* tp = &tileBuf[t & 1][6 * tid];         // this thread's 2 points
      const int li = t * TILE + 2 * tid;                  // local point index (ascending)
      float x0 = tp[0], y0 = tp[1], z0 = tp[2];
      float x1 = tp[3], y1 = tp[4], z1 = tp[5];
      float d0 = dist[li], d1 = dist[li + 1];

      float ax = x0 - qx, ay = y0 - qy, az = z0 - qz;
      float dd0 = ax * ax + ay * ay + az * az;            // contract(off): mul/add like ref
      float ux = x1 - qx, uy = y1 - qy, uz = z1 - qz;
      float dd1 = ux * ux + uy * uy + uz * uz;
      float n0 = fminf(d0, dd0);
      float n1 = fminf(d1, dd1);
      dist[li] = n0;
      dist[li + 1] = n1;
      if (n0 > bestV) { bestV = n0; bestI = li; }         // '>' keeps earliest index on ties
      if (n1 > bestV) { bestV = n1; bestI = li + 1; }
    }

    // prefetch tile 0 for the next iteration; overlaps the reduction + global sync
    issue_tile_copy(mypts, tileBuf[0], tid);

    // wave32 argmax reduction (prefer lower index on ties)
    for (int m = 16; m > 0; m >>= 1) {
      float ov = __shfl_xor(bestV, m, 32);
      int   oi = __shfl_xor(bestI, m, 32);
      if (ov > bestV || (ov == bestV && oi < bestI)) { bestV = ov; bestI = oi; }
    }
    if (lane == 0) { redV[wv] = bestV; redI[wv] = bestI; }
    __syncthreads();

    if (tid == 0) {
      float bv = redV[0]; int bi = redI[0];
      for (int j = 1; j < THREADS / 32; ++j) {
        float v = redV[j]; int ii = redI[j];
        if (v > bv || (v == bv && ii < bi)) { bv = v; bi = ii; }
      }
      __hip_atomic_store(&partV[bx], bv, __ATOMIC_RELAXED, __HIP_MEMORY_SCOPE_AGENT);
      __hip_atomic_store(&partI[bx], c * CHUNK + bi, __ATOMIC_RELAXED, __HIP_MEMORY_SCOPE_AGENT);
      __threadfence();
      int prev = __hip_atomic_fetch_add(&ctrl[b].cnt, 1, __ATOMIC_ACQ_REL, __HIP_MEMORY_SCOPE_AGENT);
      if (prev == WGPB - 1) {
        // last arriver: reduce the 8 partials, publish winner, write outputs
        float wbv = -1.0f; int wbi = 0;
        for (int j = 0; j < WGPB; ++j) {
          float v = __hip_atomic_load(&partV[b * WGPB + j], __ATOMIC_RELAXED, __HIP_MEMORY_SCOPE_AGENT);
          int  ii = __hip_atomic_load(&partI[b * WGPB + j], __ATOMIC_RELAXED, __HIP_MEMORY_SCOPE_AGENT);
          if (j == 0 || v > wbv || (v == wbv && ii < wbi)) { wbv = v; wbi = ii; }
        }
        __hip_atomic_store(&winner[b], wbi, __ATOMIC_RELAXED, __HIP_MEMORY_SCOPE_AGENT);
        outIdx[(size_t)b * NPOINTS + k] = wbi;
        const float* wp = pts + 3 * (size_t)wbi;
        float* op = outPts + ((size_t)b * NPOINTS + k) * 3;
        op[0] = wp[0]; op[1] = wp[1]; op[2] = wp[2];
        __hip_atomic_store(&ctrl[b].cnt, 0, __ATOMIC_RELAXED, __HIP_MEMORY_SCOPE_AGENT);
        __threadfence();
        __hip_atomic_fetch_add(&ctrl[b].gen, 1, __ATOMIC_ACQ_REL, __HIP_MEMORY_SCOPE_AGENT);
      } else {
        while (__hip_atomic_load(&ctrl[b].gen, __ATOMIC_ACQUIRE, __HIP_MEMORY_SCOPE_AGENT) < k)
          __builtin_amdgcn_s_sleep(1);
      }
      bcast = __hip_atomic_load(&winner[b], __ATOMIC_RELAXED, __HIP_MEMORY_SCOPE_AGENT);
    }
    __syncthreads();
    w = bcast;
  }
}

extern "C" void kernel_launch(void* const* d_in, const int* in_sizes, int n_in,
                              void* d_out, int out_size, void* d_ws, size_t ws_size,
                              hipStream_t stream) {
  (void)in_sizes; (void)n_in; (void)out_size; (void)ws_size;
  const float* inp = (const float*)d_in[0];

  float* outPts = (float*)d_out;                                  // [32,4096,3] f32
  int*   outIdx = (int*)((float*)d_out + (size_t)NBATCH * NPOINTS * 3);  // [32,4096] i32

  char*  ws     = (char*)d_ws;
  Ctrl*  ctrl   = (Ctrl*)ws;                                      // 32 * 128B
  float* partV  = (float*)(ws + 4096);                            // 256 * 4B
  int*   partI  = (int*)(ws + 4096 + 1024);                       // 256 * 4B
  int*   winner = (int*)(ws + 4096 + 2048);                       // 32 * 4B

  fps_init_kernel<<<1, 64, 0, stream>>>(ctrl, winner);
  fps_main_kernel<<<NBATCH * WGPB, THREADS, 0, stream>>>(inp, outPts, outIdx,
                                                         ctrl, partV, partI, winner);
}